// BaseModel_39745627357431
// MI455X (gfx1250) — compile-verified
//
#include <hip/hip_runtime.h>
#include <stdint.h>

typedef float v4f __attribute__((ext_vector_type(4)));

#define NB   11
#define NO   12
#define NPOS 144                    // 12*12 positions per board
#define NPAT 27
#define PER_BOARD (NPAT * NPOS)     // 3888 floats per board
#define V4_PER_BOARD (PER_BOARD/4)  // 972 float4 per board
#define BPB  4                      // boards per block (4*121 floats = 1936 B, 16B-aligned chunks)

// Padded-board lookup: P is 13x13; interior comes from LDS board B[11][11],
// top/bottom edges are +1, left/right edges are -1, corners are 0.
__device__ __forceinline__ float getP(const float* B, int r, int c) {
    if (r >= 1 && r <= NB && c >= 1 && c <= NB) return B[(r - 1) * NB + (c - 1)];
    if ((r == 0 || r == NB + 1) && c >= 1 && c <= NB) return 1.0f;
    if ((c == 0 || c == NB + 1) && r >= 1 && r <= NB) return -1.0f;
    return 0.0f; // corners
}

__global__ __launch_bounds__(256) void hex_encode_kernel(const float* __restrict__ boards,
                                                         float* __restrict__ out,
                                                         int batch) {
    __shared__ __align__(16) float B4[BPB * NB * NB];  // 484 floats, contiguous 4-board chunk
    __shared__ unsigned mask[BPB * NPOS];              // 27-bit pattern mask per position

    const int b0 = blockIdx.x * BPB;
    const int t  = threadIdx.x;
    const int nb = min(BPB, batch - b0);               // boards handled by this block

    // ---- Stage 4 boards into LDS via gfx1250 async memory->LDS path ----
    if (nb == BPB) {
        // full chunk: 1936 B, 16B-aligned (1936 = 121*16, base offset = blockIdx.x*1936)
        if (t < 121) {
            unsigned lds_byte = (unsigned)(uintptr_t)(&B4[t * 4]);
            uint64_t gaddr    = (uint64_t)(uintptr_t)(boards + (size_t)b0 * (NB * NB) + t * 4);
            asm volatile("global_load_async_to_lds_b128 %0, %1, off"
                         :: "v"(lds_byte), "v"(gaddr)
                         : "memory");
        }
    } else {
        // tail block (not hit for batch % 4 == 0): per-float async copies
        const int nflt = nb * NB * NB;
        for (int k = t; k < nflt; k += 256) {
            unsigned lds_byte = (unsigned)(uintptr_t)(&B4[k]);
            uint64_t gaddr    = (uint64_t)(uintptr_t)(boards + (size_t)b0 * (NB * NB) + k);
            asm volatile("global_load_async_to_lds_b32 %0, %1, off"
                         :: "v"(lds_byte), "v"(gaddr)
                         : "memory");
        }
    }
    asm volatile("s_wait_asynccnt 0" ::: "memory");
    __syncthreads();

    // ---- Per-position 27-bit pattern masks (fully occupied: 576 over 256 lanes) ----
    for (int k = t; k < nb * NPOS; k += 256) {
        const int board = k / NPOS;
        const int pos   = k - board * NPOS;
        const float* B  = &B4[board * NB * NB];
        const int i = pos / NO, j = pos % NO;
        float a0 = getP(B, i, j);
        float a1 = getP(B, i, j + 1);
        float a2 = getP(B, i + 1, j);
        int c0 = (int)a0 + 1;   // {-1,0,1} -> {0,1,2}, exact float values
        int c1 = (int)a1 + 1;
        int c2 = (int)a2 + 1;
        unsigned m;
        if (i == 0 && j == 0)             // eq0 forced true: p1,p2 fixed, p0 free
            m = (1u << (c1 * 3 + c2)) * ((1u << 0) | (1u << 9) | (1u << 18));
        else if (i == 0 && j == NO - 1)   // eq1 forced true: p0,p2 fixed, p1 free
            m = (1u << (c0 * 9 + c2)) * ((1u << 0) | (1u << 3) | (1u << 6));
        else if (i == NO - 1 && j == 0)   // eq2 forced true: p0,p1 fixed, p2 free
            m = (1u << (c0 * 9 + c1 * 3)) * 7u;
        else                              // exactly one pattern matches
            m = 1u << (c0 * 9 + c1 * 3 + c2);
        mask[k] = m;
    }
    __syncthreads();

    // ---- Stream 4 x 3888 contiguous floats as NT b128 stores ----
    for (int board = 0; board < nb; ++board) {
        const unsigned* M = &mask[board * NPOS];
        v4f* o = (v4f*)(out + (size_t)(b0 + board) * PER_BOARD);
#pragma unroll
        for (int q0 = 0; q0 < 4; ++q0) {
            int q = t + q0 * 256;
            if (q < V4_PER_BOARD) {
                int p  = q / 36;              // pattern index (144 floats = 36 float4 per pattern)
                int r4 = (q - p * 36) * 4;    // starting position within the 12x12 grid
                v4f v;
                v.x = (float)((M[r4 + 0] >> p) & 1u);
                v.y = (float)((M[r4 + 1] >> p) & 1u);
                v.z = (float)((M[r4 + 2] >> p) & 1u);
                v.w = (float)((M[r4 + 3] >> p) & 1u);
                __builtin_nontemporal_store(v, o + q);   // output >> L2 (510MB vs 192MB): TH=NT
            }
        }
    }
}

extern "C" void kernel_launch(void* const* d_in, const int* in_sizes, int n_in,
                              void* d_out, int out_size, void* d_ws, size_t ws_size,
                              hipStream_t stream) {
    const float* boards = (const float*)d_in[0];
    float* out = (float*)d_out;
    const int batch   = in_sizes[0] / (NB * NB);          // 32768
    const int nblocks = (batch + BPB - 1) / BPB;          // 8192
    hipLaunchKernelGGL(hex_encode_kernel, dim3(nblocks), dim3(256), 0, stream, boards, out, batch);
    (void)n_in; (void)out_size; (void)d_ws; (void)ws_size;
}